// ROIPoolingNetwork_11991548691203
// MI455X (gfx1250) — compile-verified
//
#include <hip/hip_runtime.h>
#include <hip/hip_bf16.h>
#include <stdint.h>

// Problem constants (match reference)
#define CCH      512
#define FH       64
#define FW       64
#define ROI      7
#define CPB      8              // channels per TDM tile (chunk)
#define MAXSPAN  24             // max ROI span in feature cells (exact bound)
#define NTHREADS 128            // 4 wave32s
#define KCHUNKS  8              // chunks per block -> grid.y = 512/(CPB*KCHUNKS) = 8

#define TILE_ELTS  (CPB * MAXSPAN * MAXSPAN)    // 4608 floats
#define TILE_BYTES (TILE_ELTS * 4)              // 18432 B per buffer
#define OUTS_PER_CHUNK (CPB * ROI * ROI)        // 392
#define VEC_PER_CHUNK  (OUTS_PER_CHUNK / 4)     // 98 float4 stores

typedef unsigned int u32x4 __attribute__((ext_vector_type(4)));
typedef int          i32x4 __attribute__((ext_vector_type(4)));
typedef int          i32x8 __attribute__((ext_vector_type(8)));
typedef float        f32x4 __attribute__((ext_vector_type(4)));

// Fire one TDM load: tile [CPB x hspan x wspan] fp32 from global (strides FW, FH*FW)
// into LDS at byte offset `lds`, densely packed [c][y][x].
__device__ __forceinline__ void tdm_issue(const float* gptr, uint32_t lds,
                                          int wspan, int hspan)
{
    uint64_t ga = (uint64_t)(uintptr_t)gptr;

    // D# group 0: count=1 | lds_addr | global_addr[56:0] | type=2
    u32x4 g0;
    g0[0] = 1u;
    g0[1] = lds;
    g0[2] = (uint32_t)ga;
    g0[3] = (uint32_t)((ga >> 32) & 0x01FFFFFFu) | (2u << 30);

    // D# group 1: data_size=4B | tensor_dim0/1 | tile_dim0/1/2 | dim strides
    const uint32_t td0 = FW;
    const uint32_t td1 = FH;
    const uint64_t s0  = FW;                   // row stride (elements)
    const uint64_t s1  = (uint64_t)FH * FW;    // channel stride (elements)
    i32x8 g1;
    g1[0] = (int)(2u << 16);                                     // data_size=2 -> 4B
    g1[1] = (int)((td0 & 0xFFFFu) << 16);                        // tensor_dim0[15:0]
    g1[2] = (int)((td0 >> 16) | ((td1 & 0xFFFFu) << 16));
    g1[3] = (int)((td1 >> 16) | ((uint32_t)wspan << 16));        // tile_dim0
    g1[4] = (int)((uint32_t)hspan | ((uint32_t)CPB << 16));      // tile_dim1 | tile_dim2
    g1[5] = (int)(uint32_t)s0;
    g1[6] = (int)((uint32_t)(s0 >> 32) | ((uint32_t)(s1 & 0xFFFFu) << 16));
    g1[7] = (int)(uint32_t)(s1 >> 16);

    i32x4 g2; g2[0] = CCH; g2[1] = 0; g2[2] = 0; g2[3] = 0;      // tensor_dim2
    i32x4 g3 = {0, 0, 0, 0};

#if defined(__clang_major__) && __clang_major__ >= 23
    i32x8 gx = {0, 0, 0, 0, 0, 0, 0, 0};
    __builtin_amdgcn_tensor_load_to_lds(g0, g1, g2, g3, gx, 0);
#else
    __builtin_amdgcn_tensor_load_to_lds(g0, g1, g2, g3, 0);
#endif
}

__global__ __launch_bounds__(NTHREADS)
void roi_pool_tdm_kernel(const float* __restrict__ feats,   // [512,64,64]
                         const float* __restrict__ props,   // [N,4] pixel coords
                         float* __restrict__ out)           // [N,512,7,7]
{
    __shared__ float tile[2][TILE_ELTS];
    __shared__ int   edges[ROI * ROI];      // packed hs | he<<8 | ws<<16 | we<<24

    const int n     = blockIdx.x;                       // proposal id
    const int cbase = blockIdx.y * (KCHUNKS * CPB);     // first channel of this block
    const int tid   = threadIdx.x;

    // proposal -> integer feature-map coords: floor(p/16), p >= 0 so trunc==floor.
    const float inv = 1.0f / 16.0f;
    int x1 = (int)(props[n * 4 + 0] * inv);
    int y1 = (int)(props[n * 4 + 1] * inv);
    int x2 = (int)(props[n * 4 + 2] * inv);
    int y2 = (int)(props[n * 4 + 3] * inv);
    x1 = __builtin_amdgcn_readfirstlane(x1);
    y1 = __builtin_amdgcn_readfirstlane(y1);
    x2 = __builtin_amdgcn_readfirstlane(x2);
    y2 = __builtin_amdgcn_readfirstlane(y2);

    int wspan = x2 - x1;
    int hspan = y2 - y1;
    if (wspan < 1) wspan = 1;  if (wspan > MAXSPAN) wspan = MAXSPAN;
    if (hspan < 1) hspan = 1;  if (hspan > MAXSPAN) hspan = MAXSPAN;

    const float* gbase = feats + (size_t)y1 * FW + (size_t)x1;

    // Wave 0: fire + drain the first TDM load while waves 1-2 build bin tables.
    uint32_t lds0 = 0;
    if (tid < 32) {
        lds0 = (uint32_t)(uintptr_t)(&tile[0][0]);   // low 32 bits = LDS byte offset
        tdm_issue(gbase + (size_t)cbase * (FH * FW), lds0, wspan, hspan);
        __builtin_amdgcn_s_wait_tensorcnt(0);
    } else if (tid >= 32 && tid < 32 + ROI * ROI) {
        // PyTorch AdaptiveMaxPool bin edges (relative to tile origin), packed.
        int r  = tid - 32;
        int oh = r / ROI, ow = r - oh * ROI;
        int hs = (oh * hspan) / ROI;
        int he = ((oh + 1) * hspan + (ROI - 1)) / ROI;
        int ws = (ow * wspan) / ROI;
        int we = ((ow + 1) * wspan + (ROI - 1)) / ROI;
        edges[r] = hs | (he << 8) | (ws << 16) | (we << 24);
    }
    __syncthreads();

    const int   plane = wspan * hspan;
    const float NEGF  = -__builtin_inff();
    float* outBlock = out + (size_t)n * (CCH * ROI * ROI) + (size_t)cbase * (ROI * ROI);

    for (int k = 0; k < KCHUNKS; ++k) {
        const int cur = k & 1;

        // Prefetch next chunk into the other buffer (overlaps with compute below).
        if (tid < 32 && (k + 1) < KCHUNKS) {
            tdm_issue(gbase + (size_t)(cbase + (k + 1) * CPB) * (FH * FW),
                      lds0 + (uint32_t)((cur ^ 1) * TILE_BYTES), wspan, hspan);
        }

        // Compute 392 outputs of this chunk from LDS; 4 per thread, b128 NT store.
        if (tid < VEC_PER_CHUNK) {
            const float* buf = &tile[cur][0];
            f32x4 v;
#pragma unroll
            for (int j = 0; j < 4; ++j) {
                int idx = 4 * tid + j;                 // 0..391
                int c   = idx / (ROI * ROI);
                int r   = idx - c * (ROI * ROI);
                int e   = edges[r];
                int hs = e & 0xFF, he = (e >> 8) & 0xFF;
                int ws = (e >> 16) & 0xFF, we = (e >> 24) & 0xFF;
                const float* base = buf + c * plane;
                float m = NEGF;
                for (int y = hs; y < he; ++y) {
                    const float* row = base + y * wspan;
                    for (int x = ws; x < we; ++x) m = fmaxf(m, row[x]);
                }
                v[j] = m;
            }
            // Stream the output: keep the 192MB L2 for the feature map, not the result.
            __builtin_nontemporal_store(
                v, (f32x4*)(outBlock + (size_t)k * OUTS_PER_CHUNK + 4 * tid));
        }

        __syncthreads();                               // all readers of buf[cur] done
        if (tid < 32 && (k + 1) < KCHUNKS)
            __builtin_amdgcn_s_wait_tensorcnt(0);      // next buffer filled
        __syncthreads();                               // publish to all waves
    }
}

extern "C" void kernel_launch(void* const* d_in, const int* in_sizes, int n_in,
                              void* d_out, int out_size, void* d_ws, size_t ws_size,
                              hipStream_t stream) {
    const float* feats = (const float*)d_in[0];   // [1,512,64,64] fp32
    const float* props = (const float*)d_in[1];   // [N,4] fp32
    float* out = (float*)d_out;                   // [N,512,7,7] fp32

    const int nprop = in_sizes[1] / 4;
    dim3 grid((unsigned)nprop, CCH / (CPB * KCHUNKS));   // 2000 x 8 blocks
    roi_pool_tdm_kernel<<<grid, NTHREADS, 0, stream>>>(feats, props, out);
}